// WSRRR_5282809774230
// MI455X (gfx1250) — compile-verified
//
#include <hip/hip_runtime.h>
#include <math.h>

typedef float v2f __attribute__((ext_vector_type(2)));
typedef float v8f __attribute__((ext_vector_type(8)));

#define N_PIX 65536
#define CCH   200
#define CPAD  208
#define NWG_GEMM 512
#define CDIV(a,b) (((a)+(b)-1)/(b))

__constant__ float c_lo[10] = {
  0.003335725285001549f, -0.012580751999015526f, -0.006241490213011705f,
  0.07757149384006515f, -0.03224486958502952f, -0.24229488706619015f,
  0.13842814590110342f, 0.7243085284385744f, 0.6038292697974729f,
  0.160102397974125f };
__constant__ float c_hi[10] = {
  0.160102397974125f, -0.6038292697974729f, 0.7243085284385744f,
  -0.13842814590110342f, -0.24229488706619015f, 0.03224486958502952f,
  0.07757149384006515f, 0.006241490213011705f, -0.012580751999015526f,
  -0.003335725285001549f };

// Mallat-layout position for to_2d/from_2d row index n (3 levels, 256x256).
__device__ __forceinline__ int map_n(int n) {
  if (n < 1024) { return (n >> 5) * 256 + (n & 31); }         // LL 32x32
  int base, lvl;
  if (n < 4096)       { base = 1024;  lvl = 0; }
  else if (n < 16384) { base = 4096;  lvl = 1; }
  else                { base = 16384; lvl = 2; }
  int side = 32 << lvl;
  int r = n - base, ss = side * side;
  int b = r / ss, t = r % ss;
  int i = t / side, j = t % side;
  if (b == 0) return i * 256 + (side + j);            // lh: lowH highW
  if (b == 1) return (side + i) * 256 + j;            // hl: highH lowW
  return (side + i) * 256 + (side + j);               // hh
}

__global__ __launch_bounds__(256) void k_zero(float* __restrict__ p, int n) {
  int i = blockIdx.x * 256 + threadIdx.x;
  if (i < n) p[i] = 0.f;
}

__global__ __launch_bounds__(256) void k_x_to_planar(const float* __restrict__ x,
                                                     float* __restrict__ buf) {
  int id = blockIdx.x * 256 + threadIdx.x;
  if (id >= CCH * N_PIX) return;
  int c = id / N_PIX, n = id % N_PIX;
  buf[(size_t)c * N_PIX + n] = x[(size_t)n * CCH + c];
}

// Forward DWT along W: in[c][i][j], j<s -> low at out[..][jo], high at out[..][h+jo]
__global__ __launch_bounds__(256) void k_dwt_w(const float* __restrict__ in,
                                               float* __restrict__ out, int s, int C) {
  int h = s >> 1;
  int id = blockIdx.x * 256 + threadIdx.x;
  if (id >= C * s * h) return;
  int c = id / (s * h), rem = id % (s * h);
  int i = rem / h, jo = rem % h;
  const float* row = in + (size_t)c * N_PIX + i * 256;
  float a = 0.f, d = 0.f;
  #pragma unroll
  for (int k = 0; k < 10; ++k) {
    int j = (2 * jo + 1 - k + s) % s;
    float v = row[j];
    a += v * c_lo[k];
    d += v * c_hi[k];
  }
  float* orow = out + (size_t)c * N_PIX + i * 256;
  orow[jo] = a; orow[h + jo] = d;
}

__global__ __launch_bounds__(256) void k_dwt_h(const float* __restrict__ in,
                                               float* __restrict__ out, int s, int C) {
  int h = s >> 1;
  int id = blockIdx.x * 256 + threadIdx.x;
  if (id >= C * h * s) return;
  int c = id / (h * s), rem = id % (h * s);
  int io = rem / s, j = rem % s;
  const float* pl = in + (size_t)c * N_PIX;
  float a = 0.f, d = 0.f;
  #pragma unroll
  for (int k = 0; k < 10; ++k) {
    int i = (2 * io + 1 - k + s) % s;
    float v = pl[i * 256 + j];
    a += v * c_lo[k];
    d += v * c_hi[k];
  }
  float* po = out + (size_t)c * N_PIX;
  po[io * 256 + j] = a; po[(h + io) * 256 + j] = d;
}

// Inverse DWT along H: a = rows[0:h], d = rows[h:s] (any column)
__global__ __launch_bounds__(256) void k_idwt_h(const float* __restrict__ in,
                                                float* __restrict__ out, int s, int C) {
  int h = s >> 1;
  int id = blockIdx.x * 256 + threadIdx.x;
  if (id >= C * h * s) return;
  int c = id / (h * s), rem = id % (h * s);
  int io = rem / s, j = rem % s;
  const float* pl = in + (size_t)c * N_PIX;
  float e = 0.f, o = 0.f;
  #pragma unroll
  for (int k = 0; k < 5; ++k) {
    int idx = (io + k) % h;
    float a = pl[idx * 256 + j];
    float d = pl[(h + idx) * 256 + j];
    e += a * c_lo[2 * k + 1] + d * c_hi[2 * k + 1];
    o += a * c_lo[2 * k]     + d * c_hi[2 * k];
  }
  float* po = out + (size_t)c * N_PIX;
  po[(2 * io) * 256 + j] = e; po[(2 * io + 1) * 256 + j] = o;
}

__global__ __launch_bounds__(256) void k_idwt_w(const float* __restrict__ in,
                                                float* __restrict__ out, int s, int C) {
  int h = s >> 1;
  int id = blockIdx.x * 256 + threadIdx.x;
  if (id >= C * s * h) return;
  int c = id / (s * h), rem = id % (s * h);
  int i = rem / h, jo = rem % h;
  const float* row = in + (size_t)c * N_PIX + i * 256;
  float e = 0.f, o = 0.f;
  #pragma unroll
  for (int k = 0; k < 5; ++k) {
    int idx = (jo + k) % h;
    float a = row[idx], d = row[h + idx];
    e += a * c_lo[2 * k + 1] + d * c_hi[2 * k + 1];
    o += a * c_lo[2 * k]     + d * c_hi[2 * k];
  }
  float* orow = out + (size_t)c * N_PIX + i * 256;
  orow[2 * jo] = e; orow[2 * jo + 1] = o;
}

__global__ __launch_bounds__(256) void k_to2d(const float* __restrict__ buf,
                                              float* __restrict__ wy) {
  long id = (long)blockIdx.x * 256 + threadIdx.x;
  if (id >= (long)N_PIX * CPAD) return;
  int n = (int)(id / CPAD), c = (int)(id % CPAD);
  wy[id] = (c < CCH) ? buf[(size_t)c * N_PIX + map_n(n)] : 0.f;
}

__global__ __launch_bounds__(256) void k_from2d(const float* __restrict__ src,
                                                float* __restrict__ dst,
                                                int C, int cstride) {
  long id = (long)blockIdx.x * 256 + threadIdx.x;
  if (id >= (long)N_PIX * C) return;
  int n = (int)(id / C), c = (int)(id % C);
  dst[(size_t)c * N_PIX + map_n(n)] = src[(size_t)n * cstride + c];
}

// Exact per-channel median of |wy[16384:,c]| via 8-bit radix select (4 rounds).
__global__ __launch_bounds__(256) void k_median(const float* __restrict__ wy,
                                                float* __restrict__ omega) {
  __shared__ unsigned hist[256];
  __shared__ unsigned s_prefix;
  __shared__ int s_k;
  int c = blockIdx.x, tid = threadIdx.x;
  float vals[2];
  for (int sel = 0; sel < 2; ++sel) {
    if (tid == 0) { s_k = 24575 + sel; s_prefix = 0u; }
    __syncthreads();
    unsigned mask = 0u;
    for (int round = 0; round < 4; ++round) {
      int shift = 24 - 8 * round;
      hist[tid] = 0u;
      __syncthreads();
      unsigned pref = s_prefix;
      for (int n = 16384 + tid; n < 65536; n += 256) {
        unsigned key = __float_as_uint(fabsf(wy[(size_t)n * CPAD + c]));
        if ((key & mask) == pref) atomicAdd(&hist[(key >> shift) & 255u], 1u);
      }
      __syncthreads();
      if (tid == 0) {
        int kk = s_k; unsigned cum = 0; int bsel = 255;
        for (int b = 0; b < 256; ++b) {
          unsigned hc = hist[b];
          if (cum + hc > (unsigned)kk) { bsel = b; break; }
          cum += hc;
        }
        s_k = kk - (int)cum;
        s_prefix = pref | ((unsigned)bsel << shift);
      }
      __syncthreads();
      mask |= (255u << shift);
    }
    vals[sel] = __uint_as_float(s_prefix);
    __syncthreads();
  }
  if (tid == 0) omega[c] = 0.5f * (vals[0] + vals[1]) / 0.6745f + 1e-30f;
}

__global__ __launch_bounds__(256) void k_scale_wy(float* __restrict__ wy,
                                                  const float* __restrict__ omega) {
  long id = (long)blockIdx.x * 256 + threadIdx.x;
  if (id >= (long)N_PIX * CCH) return;
  int n = (int)(id / CCH), c = (int)(id % CCH);
  wy[(size_t)n * CPAD + c] /= omega[c];
}

// Gram of raw x: GX[c1][c2] += sum_n x[n,c1]*x[n,c2]
__global__ __launch_bounds__(256) void k_gram(const float* __restrict__ x,
                                              float* __restrict__ gx) {
  __shared__ float sX[256];
  int ti = blockIdx.x, tj = blockIdx.y, z = blockIdx.z;
  int tid = threadIdx.x;
  int ci = tid >> 4, cj = tid & 15;
  int nl = tid >> 5, cl = tid & 31;
  int cglob = (cl < 16) ? (ti * 16 + cl) : (tj * 16 + cl - 16);
  float acc = 0.f;
  for (int nb = z * 4096; nb < (z + 1) * 4096; nb += 8) {
    sX[tid] = (cglob < CCH) ? x[(size_t)(nb + nl) * CCH + cglob] : 0.f;
    __syncthreads();
    #pragma unroll
    for (int e = 0; e < 8; ++e)
      acc += sX[e * 32 + ci] * sX[e * 32 + 16 + cj];
    __syncthreads();
  }
  atomicAdd(&gx[(size_t)(ti * 16 + ci) * CPAD + tj * 16 + cj], acc);
}

__global__ __launch_bounds__(256) void k_formG(const float* __restrict__ gx,
                                               const float* __restrict__ omega,
                                               float* __restrict__ g) {
  int id = blockIdx.x * 256 + threadIdx.x;
  if (id >= 200 * 200) return;
  int c1 = id / 200, c2 = id % 200;
  g[id] = gx[(size_t)c1 * CPAD + c2] / (omega[c1] * omega[c2]);
}

// 200x200 cyclic Jacobi eigensolver (one workgroup), sorts descending, writes
// v0[i][j] = Vg[j, sorted_i]  (matches vh[:, :11] of the reference SVD).
__global__ __launch_bounds__(256) void k_jacobi200(float* __restrict__ G,
                                                   float* __restrict__ Q,
                                                   float* __restrict__ vmat,
                                                   int* __restrict__ idxg) {
  __shared__ float shc, shs;
  __shared__ float ev[200];
  __shared__ int taken[200];
  int tid = threadIdx.x;
  for (int i = tid; i < 200 * 200; i += 256) Q[i] = ((i / 200) == (i % 200)) ? 1.f : 0.f;
  __threadfence(); __syncthreads();
  for (int sweep = 0; sweep < 6; ++sweep) {
    for (int p = 0; p < 199; ++p) {
      for (int q = p + 1; q < 200; ++q) {
        if (tid == 0) {
          float apq = G[p * 200 + q];
          if (fabsf(apq) > 1e-12f) {
            float app = G[p * 200 + p], aqq = G[q * 200 + q];
            float tau = (aqq - app) / (2.f * apq);
            float t = ((tau >= 0.f) ? 1.f : -1.f) / (fabsf(tau) + sqrtf(1.f + tau * tau));
            float cc = 1.f / sqrtf(1.f + t * t);
            shc = cc; shs = t * cc;
          } else { shc = 1.f; shs = 0.f; }
        }
        __syncthreads();
        float cc = shc, ss = shs;
        if (ss != 0.f) {
          for (int k = tid; k < 200; k += 256) {
            float gp = G[p * 200 + k], gq = G[q * 200 + k];
            G[p * 200 + k] = cc * gp - ss * gq;
            G[q * 200 + k] = ss * gp + cc * gq;
          }
          __threadfence();
        }
        __syncthreads();
        if (ss != 0.f) {
          for (int k = tid; k < 200; k += 256) {
            float gp = G[k * 200 + p], gq = G[k * 200 + q];
            G[k * 200 + p] = cc * gp - ss * gq;
            G[k * 200 + q] = ss * gp + cc * gq;
            float qp = Q[k * 200 + p], qq = Q[k * 200 + q];
            Q[k * 200 + p] = cc * qp - ss * qq;
            Q[k * 200 + q] = ss * qp + cc * qq;
          }
          __threadfence();
        }
        __syncthreads();
      }
    }
  }
  if (tid == 0) {
    for (int i = 0; i < 200; ++i) { ev[i] = G[i * 200 + i]; taken[i] = 0; }
    for (int r = 0; r < 200; ++r) {
      int best = -1; float bv = 0.f;
      for (int i = 0; i < 200; ++i)
        if (!taken[i] && (best < 0 || ev[i] > bv)) { bv = ev[i]; best = i; }
      taken[best] = 1; idxg[r] = best;
    }
  }
  __threadfence(); __syncthreads();
  for (int i = tid; i < CPAD * 16; i += 256) vmat[i] = 0.f;
  __threadfence(); __syncthreads();
  for (int f = tid; f < 200 * 11; f += 256) {
    int i = f / 11, j = f % 11;
    vmat[i * 16 + j] = Q[j * 200 + idxg[i]];
  }
}

// w0 = wy @ v (raw, for SURE threshold search)
__global__ __launch_bounds__(256) void k_gemm_w0(const float* __restrict__ wy,
                                                 const float* __restrict__ vmat,
                                                 float* __restrict__ wx) {
  long id = (long)blockIdx.x * 256 + threadIdx.x;
  if (id >= (long)N_PIX * 16) return;
  int n = (int)(id >> 4), j = (int)(id & 15);
  const float* row = wy + (size_t)n * CPAD;
  float acc = 0.f;
  for (int c = 0; c < CCH; ++c) acc += row[c] * vmat[c * 16 + j];
  wx[id] = acc;
}

// SURE threshold: one WG per (column j, segment s)
__global__ __launch_bounds__(256) void k_sure(const float* __restrict__ w0,
                                              float* __restrict__ thr) {
  __shared__ float ssum[15];
  __shared__ int scnt[15];
  int j = blockIdx.x, s = blockIdx.y, tid = threadIdx.x;
  int a, b;
  if (s == 0)      { a = 0;     b = 1024;  }
  else if (s == 1) { a = 1024;  b = 4096;  }
  else if (s == 2) { a = 4096;  b = 16384; }
  else             { a = 16384; b = 65536; }
  int M = b - a;
  float tmax = sqrtf(logf((float)M));
  float step = tmax / 14.f;
  if (tid < 15) { ssum[tid] = 0.f; scnt[tid] = 0; }
  __syncthreads();
  float ls[15]; int lc[15];
  #pragma unroll
  for (int m = 0; m < 15; ++m) { ls[m] = 0.f; lc[m] = 0; }
  for (int n = a + tid; n < b; n += 256) {
    float xv = w0[(size_t)n * 16 + j];
    float ax = fabsf(xv), x2 = xv * xv;
    #pragma unroll
    for (int m = 0; m < 15; ++m) {
      float t = step * m;
      if (ax > t) { ls[m] += t * t; lc[m] += 1; }
      else        { ls[m] += x2 - t * t; }
    }
  }
  #pragma unroll
  for (int m = 0; m < 15; ++m) { atomicAdd(&ssum[m], ls[m]); atomicAdd(&scnt[m], lc[m]); }
  __syncthreads();
  if (tid == 0) {
    int best = 0; float bv = 3.4e38f;
    for (int m = 0; m < 15; ++m) {
      float sure = (float)M + ssum[m] - 2.f * (float)scnt[m];
      if (sure < bv) { bv = sure; best = m; }
    }
    int row = (s == 0) ? 0 : s;   // row0 = thr_low, rows1..3 = thr_det[0..2]
    thr[row * 16 + j] = step * best;
  }
}

// Fused per-iteration kernel: wx = soft(wy @ v, thr); Bpart += wx^T wy.
// 8 wave32 per WG; each wave owns one 16-row block; both GEMMs on WMMA f32 16x16x4.
__global__ __launch_bounds__(256) void k_fused(const float* __restrict__ wy,
                                               const float* __restrict__ vmat,
                                               const float* __restrict__ thr,
                                               float* __restrict__ wx,
                                               float* __restrict__ bpart,
                                               int iter0) {
  __shared__ float sV[CPAD * 16];   // v matrix (K-major)
  __shared__ float sB[16 * CPAD];   // per-WG B tile accumulator
  __shared__ float sW[8 * 256];     // per-wave wx tile (row-major 16x16)
  int tid = threadIdx.x;
  int wave = tid >> 5, lane = tid & 31;
  int g = blockIdx.x;
  for (int i = tid; i < CPAD * 16; i += 256) { sV[i] = vmat[i]; sB[i] = 0.f; }
  __syncthreads();

  int n0 = (g * 8 + wave) * 16;
  int m_lane = lane & 15;
  int koff   = (lane >> 4) * 2;   // K offset for A/B fragments (lanes 16-31 -> K+2)
  int m_base = (lane >> 4) * 8;   // D row offset for upper lane half

  int seg  = (n0 < 4096) ? 0 : ((n0 < 16384) ? 1 : 2);
  int trow = (iter0 && n0 < 1024) ? 0 : (1 + seg);
  float tj = thr[trow * 16 + m_lane];

  const float* wyA = wy + (size_t)(n0 + m_lane) * CPAD;
  __builtin_prefetch(wyA, 0, 1);

  // GEMM 1: D(16x16) = wy_block(16x208) x v(208x16)
  v8f acc = {};
  for (int k4 = 0; k4 < CPAD; k4 += 4) {
    v2f a, b;
    a.x = wyA[k4 + koff];
    a.y = wyA[k4 + koff + 1];
    b.x = sV[(k4 + koff) * 16 + m_lane];
    b.y = sV[(k4 + koff + 1) * 16 + m_lane];
    acc = __builtin_amdgcn_wmma_f32_16x16x4_f32(false, a, false, b, (short)0, acc,
                                                false, false);
  }

  // soft-threshold, stage tile to LDS and store wx
  float* sWp = sW + wave * 256;
  #pragma unroll
  for (int r = 0; r < 8; ++r) {
    float w = acc[r];
    float mag = fabsf(w) - tj;
    float sv = (mag > 0.f) ? copysignf(mag, w) : 0.f;
    int m = m_base + r;
    sWp[m * 16 + m_lane] = sv;
    wx[(size_t)(n0 + m) * 16 + m_lane] = sv;
  }

  // GEMM 2: B_tile(16x208) += wx_block^T(16x16) x wy_block(16x208)
  for (int ct = 0; ct < 13; ++ct) {
    v8f accb = {};
    #pragma unroll
    for (int kk = 0; kk < 16; kk += 4) {
      v2f a, b;
      a.x = sWp[(kk + koff) * 16 + m_lane];          // A'[j][row]
      a.y = sWp[(kk + koff + 1) * 16 + m_lane];
      const float* wr = wy + (size_t)(n0 + kk + koff) * CPAD + ct * 16 + m_lane;
      b.x = wr[0];                                    // B'[row][c]
      b.y = wr[CPAD];
      accb = __builtin_amdgcn_wmma_f32_16x16x4_f32(false, a, false, b, (short)0, accb,
                                                   false, false);
    }
    #pragma unroll
    for (int r = 0; r < 8; ++r)
      atomicAdd(&sB[(m_base + r) * CPAD + ct * 16 + m_lane], accb[r]);
  }
  __syncthreads();
  float* bp = bpart + (size_t)g * (16 * CPAD);
  for (int i = tid; i < 16 * CPAD; i += 256) bp[i] = sB[i];
}

__global__ __launch_bounds__(256) void k_reduceB(const float* __restrict__ bpart,
                                                 float* __restrict__ b) {
  int i = blockIdx.x * 256 + threadIdx.x;
  if (i >= 16 * CPAD) return;
  float acc = 0.f;
  for (int g = 0; g < NWG_GEMM; ++g) acc += bpart[(size_t)g * (16 * CPAD) + i];
  b[i] = acc;
}

// Polar update: v = B^T (B B^T)^{-1/2}  via 16x16 Jacobi eigensolve.
__global__ __launch_bounds__(256) void k_polar(const float* __restrict__ b,
                                               float* __restrict__ vmat) {
  __shared__ float sG[256], sQ[256], sP[256], sInv[16];
  int tid = threadIdx.x;
  {
    int k = tid >> 4, l = tid & 15;
    float acc = 0.f;
    for (int c = 0; c < CPAD; ++c) acc += b[k * CPAD + c] * b[l * CPAD + c];
    sG[tid] = acc;
  }
  __syncthreads();
  if (tid == 0) {
    for (int i = 0; i < 256; ++i) sQ[i] = ((i >> 4) == (i & 15)) ? 1.f : 0.f;
    for (int sweep = 0; sweep < 15; ++sweep) {
      for (int p = 0; p < 15; ++p)
        for (int q = p + 1; q < 16; ++q) {
          float apq = sG[p * 16 + q];
          if (fabsf(apq) < 1e-18f) continue;
          float app = sG[p * 16 + p], aqq = sG[q * 16 + q];
          float tau = (aqq - app) / (2.f * apq);
          float t = ((tau >= 0.f) ? 1.f : -1.f) / (fabsf(tau) + sqrtf(1.f + tau * tau));
          float cc = 1.f / sqrtf(1.f + t * t), ss = t * cc;
          for (int k = 0; k < 16; ++k) {
            float gp = sG[p * 16 + k], gq = sG[q * 16 + k];
            sG[p * 16 + k] = cc * gp - ss * gq;
            sG[q * 16 + k] = ss * gp + cc * gq;
          }
          for (int k = 0; k < 16; ++k) {
            float gp = sG[k * 16 + p], gq = sG[k * 16 + q];
            sG[k * 16 + p] = cc * gp - ss * gq;
            sG[k * 16 + q] = ss * gp + cc * gq;
            float qp = sQ[k * 16 + p], qq = sQ[k * 16 + q];
            sQ[k * 16 + p] = cc * qp - ss * qq;
            sQ[k * 16 + q] = ss * qp + cc * qq;
          }
        }
    }
    float maxl = 0.f;
    for (int e = 0; e < 16; ++e) { float l = sG[e * 16 + e]; if (l > maxl) maxl = l; }
    for (int e = 0; e < 16; ++e) {
      float l = sG[e * 16 + e];
      sInv[e] = (l > maxl * 1e-10f && l > 0.f) ? rsqrtf(l) : 0.f;
    }
  }
  __syncthreads();
  {
    int k = tid >> 4, j = tid & 15;
    float acc = 0.f;
    for (int e = 0; e < 16; ++e) acc += sQ[k * 16 + e] * sInv[e] * sQ[j * 16 + e];
    sP[tid] = acc;
  }
  __syncthreads();
  for (int idx = tid; idx < CPAD * 16; idx += 256) {
    int c = idx >> 4, j = idx & 15;
    float acc = 0.f;
    #pragma unroll
    for (int k = 0; k < 16; ++k) acc += b[k * CPAD + c] * sP[k * 16 + j];
    vmat[idx] = acc;
  }
}

// prod = wx @ v.T : out[n][c] = sum_j wx[n][j] * v[c][j]
__global__ __launch_bounds__(256) void k_gemm_out(const float* __restrict__ wx,
                                                  const float* __restrict__ vmat,
                                                  float* __restrict__ prod) {
  long id = (long)blockIdx.x * 256 + threadIdx.x;
  if (id >= (long)N_PIX * CCH) return;
  int n = (int)(id / CCH), c = (int)(id % CCH);
  const float* wr = wx + (size_t)n * 16;
  const float* vr = vmat + (size_t)c * 16;
  float acc = 0.f;
  #pragma unroll
  for (int j = 0; j < 11; ++j) acc += wr[j] * vr[j];
  prod[id] = acc;
}

__global__ __launch_bounds__(256) void k_emit_xx(const float* __restrict__ buf,
                                                 const float* __restrict__ omega,
                                                 float* __restrict__ out) {
  long id = (long)blockIdx.x * 256 + threadIdx.x;
  if (id >= (long)N_PIX * CCH) return;
  int n = (int)(id / CCH), c = (int)(id % CCH);
  out[id] = omega[c] * buf[(size_t)c * N_PIX + n];
}

__global__ __launch_bounds__(256) void k_emit_pcs(const float* __restrict__ bufp,
                                                  float* __restrict__ out) {
  long id = (long)blockIdx.x * 256 + threadIdx.x;
  if (id >= (long)N_PIX * 11) return;
  int n = (int)(id / 11), c = (int)(id % 11);
  out[id] = bufp[(size_t)c * N_PIX + n];
}

extern "C" void kernel_launch(void* const* d_in, const int* in_sizes, int n_in,
                              void* d_out, int out_size, void* d_ws, size_t ws_size,
                              hipStream_t stream) {
  (void)in_sizes; (void)n_in; (void)out_size; (void)ws_size;
  const float* x = (const float*)d_in[0];
  float* out = (float*)d_out;
  float* ws = (float*)d_ws;

  size_t off = 0;
  float* WY    = ws + off; off += (size_t)N_PIX * CPAD;       // padded 2D wavelet matrix
  float* BUFA  = ws + off; off += (size_t)CCH * N_PIX;        // Mallat working buffer
  float* TMP   = ws + off; off += (size_t)CCH * N_PIX;        // ping-pong / products
  float* WX    = ws + off; off += (size_t)N_PIX * 16;         // wx (cols padded to 16)
  float* BPART = ws + off; off += (size_t)NWG_GEMM * 16 * CPAD;
  float* VMAT  = ws + off; off += CPAD * 16;
  float* BMAT  = ws + off; off += 16 * CPAD;
  float* GX    = ws + off; off += CPAD * CPAD;
  float* GM    = ws + off; off += 200 * 200;
  float* QM    = ws + off; off += 200 * 200;
  float* OMEGA = ws + off; off += 256;
  float* THR   = ws + off; off += 64;
  int*   IDX   = (int*)(ws + off); off += 256;
  float* BUFP  = ws + off; off += (size_t)11 * N_PIX;

  dim3 B(256);
  // ---- forward DWT of x (channel-planar), 3 levels ----
  k_x_to_planar<<<CDIV(CCH * N_PIX, 256), B, 0, stream>>>(x, BUFA);
  for (int lvl = 0; lvl < 3; ++lvl) {
    int s = 256 >> lvl, h = s >> 1;
    k_dwt_w<<<CDIV(CCH * s * h, 256), B, 0, stream>>>(BUFA, TMP, s, CCH);
    k_dwt_h<<<CDIV(CCH * h * s, 256), B, 0, stream>>>(TMP, BUFA, s, CCH);
  }
  k_to2d<<<(int)CDIV((long)N_PIX * CPAD, 256), B, 0, stream>>>(BUFA, WY);
  // ---- omega (median of finest-level |coeffs|), wy = v2d / omega ----
  k_median<<<CCH, B, 0, stream>>>(WY, OMEGA);
  k_scale_wy<<<(int)CDIV((long)N_PIX * CCH, 256), B, 0, stream>>>(WY, OMEGA);
  // ---- initial v from SVD of inp: Gram of x scaled by omega, 200x200 Jacobi ----
  k_zero<<<CDIV(CPAD * CPAD, 256), B, 0, stream>>>(GX, CPAD * CPAD);
  k_gram<<<dim3(13, 13, 16), B, 0, stream>>>(x, GX);
  k_formG<<<CDIV(200 * 200, 256), B, 0, stream>>>(GX, OMEGA, GM);
  k_jacobi200<<<1, B, 0, stream>>>(GM, QM, VMAT, IDX);
  // ---- SURE thresholds on w0 = wy @ v ----
  k_gemm_w0<<<(int)CDIV((long)N_PIX * 16, 256), B, 0, stream>>>(WY, VMAT, WX);
  k_zero<<<1, B, 0, stream>>>(THR, 64);
  k_sure<<<dim3(11, 4), B, 0, stream>>>(WX, THR);
  // ---- 200 alternating rounds (1 thr0 + 198 scan + 1 final), each fused ----
  for (int it = 0; it < 200; ++it) {
    k_fused<<<NWG_GEMM, B, 0, stream>>>(WY, VMAT, THR, WX, BPART, (it == 0) ? 1 : 0);
    k_reduceB<<<CDIV(16 * CPAD, 256), B, 0, stream>>>(BPART, BMAT);
    k_polar<<<1, B, 0, stream>>>(BMAT, VMAT);
  }
  // ---- xx = omega * idwt(from_2d(wx @ v.T)) ----
  k_gemm_out<<<(int)CDIV((long)N_PIX * CCH, 256), B, 0, stream>>>(WX, VMAT, TMP);
  k_from2d<<<(int)CDIV((long)N_PIX * CCH, 256), B, 0, stream>>>(TMP, BUFA, CCH, CCH);
  for (int s = 64; s <= 256; s <<= 1) {
    int h = s >> 1;
    k_idwt_h<<<CDIV(CCH * h * s, 256), B, 0, stream>>>(BUFA, TMP, s, CCH);
    k_idwt_w<<<CDIV(CCH * s * h, 256), B, 0, stream>>>(TMP, BUFA, s, CCH);
  }
  k_emit_xx<<<(int)CDIV((long)N_PIX * CCH, 256), B, 0, stream>>>(BUFA, OMEGA, out);
  // ---- pcs = idwt(from_2d(wx)) ----
  k_from2d<<<(int)CDIV((long)N_PIX * 11, 256), B, 0, stream>>>(WX, BUFP, 11, 16);
  for (int s = 64; s <= 256; s <<= 1) {
    int h = s >> 1;
    k_idwt_h<<<CDIV(11 * h * s, 256), B, 0, stream>>>(BUFP, TMP, s, 11);
    k_idwt_w<<<CDIV(11 * s * h, 256), B, 0, stream>>>(TMP, BUFP, s, 11);
  }
  k_emit_pcs<<<(int)CDIV((long)N_PIX * 11, 256), B, 0, stream>>>(
      BUFP, out + (size_t)N_PIX * CCH);
}